// GraphSAGE_13529146982817
// MI455X (gfx1250) — compile-verified
//
#include <hip/hip_runtime.h>
#include <hip/hip_bf16.h>

// ---------------------------------------------------------------------------
// GraphSAGE (2-layer, mean aggregation) for MI455X / gfx1250, wave32.
//
// Math identity exploited: mean-aggregate commutes with the linear layer:
//     lin_l(mean_j x_j) = mean_j lin_l(x_j)
// => transform first with WMMA (narrow output: 16 / 8 features), then do the
// irregular edge gather/scatter on the *narrow* features. Edge traffic drops
// 8x for layer 1 and stays L2-resident (192MB L2 >> ~60MB working set).
// Dense FLOPs are trivial (~0.44 GFLOP) so we keep full f32 precision via
// V_WMMA_F32_16X16X4_F32.
// ---------------------------------------------------------------------------

#define N_NODES   50000
#define N_EDGES   600000
#define D_IN      128
#define D_HID     16
#define N_CLASSES 8
#define NTILES    ((N_NODES + 15) / 16)   // 3125 16-row WMMA tiles

typedef __attribute__((ext_vector_type(2))) float v2f;
typedef __attribute__((ext_vector_type(8))) float v8f;

// ---------------- utility: zero a float range -----------------------------
__global__ void sage_zero_kernel(float* __restrict__ p, int n) {
    int i = blockIdx.x * blockDim.x + threadIdx.x;
    if (i < n) p[i] = 0.0f;
}

// ---------------- in-degree counts (float, exact up to 2^24) --------------
__global__ void sage_degree_kernel(const long long* __restrict__ dst,
                                   float* __restrict__ cnt) {
    int e = blockIdx.x * blockDim.x + threadIdx.x;
    if (e < N_EDGES) atomicAdd(&cnt[(int)dst[e]], 1.0f);
}

// ---------------- layer-1 GEMM: xl = x@W1l, xr = x@W1r (WMMA f32) ---------
// One wave per 16-row tile; 8 waves / 256-thread block (WGP-friendly).
// A-frag (16x4 f32): lanes 0-15 hold K=khalf..khalf+1 of rows M=lane,
// lanes 16-31 hold K=2..3 (ISA 7.12.2). B-frag mirrors the K split.
__global__ __launch_bounds__(256)
void sage_gemm1_kernel(const float* __restrict__ x,
                       const float* __restrict__ W1l,
                       const float* __restrict__ W1r,
                       float* __restrict__ xl,
                       float* __restrict__ xr) {
    __shared__ float lwl[D_IN * D_HID];   // 8KB
    __shared__ float lwr[D_IN * D_HID];   // 8KB  (320KB/WGP available)
    for (int i = threadIdx.x; i < D_IN * D_HID; i += 256) {
        lwl[i] = W1l[i];
        lwr[i] = W1r[i];
    }
    __syncthreads();

    const int lane  = threadIdx.x & 31;
    const int wave  = threadIdx.x >> 5;
    const int tile  = blockIdx.x * 8 + wave;
    if (tile >= NTILES) return;           // whole wave exits together (EXEC all-1 for WMMA)

    const int col   = lane & 15;
    const int khalf = (lane >> 4) << 1;   // 0 (lanes 0-15) or 2 (lanes 16-31)

    int arow = tile * 16 + col;           // A-matrix row this lane feeds
    if (arow >= N_NODES) arow = N_NODES - 1;   // clamp (stores are guarded)
    const float* xrow = x + (size_t)arow * D_IN;

    v8f accl = {0.f,0.f,0.f,0.f,0.f,0.f,0.f,0.f};
    v8f accr = {0.f,0.f,0.f,0.f,0.f,0.f,0.f,0.f};

    for (int k = 0; k < D_IN; k += 4) {
        const int kk = k + khalf;
        v2f a;  a.x  = xrow[kk];                  a.y  = xrow[kk + 1];
        v2f bl; bl.x = lwl[kk * D_HID + col];     bl.y = lwl[(kk + 1) * D_HID + col];
        v2f br; br.x = lwr[kk * D_HID + col];     br.y = lwr[(kk + 1) * D_HID + col];
        accl = __builtin_amdgcn_wmma_f32_16x16x4_f32(false, a, false, bl,
                                                     (short)0, accl, false, false);
        accr = __builtin_amdgcn_wmma_f32_16x16x4_f32(false, a, false, br,
                                                     (short)0, accr, false, false);
    }

    // C/D layout: VGPR r <-> row r (lanes 0-15) / row r+8 (lanes 16-31).
    const int half8 = (lane >> 4) << 3;
#pragma unroll
    for (int r = 0; r < 8; ++r) {
        const int orow = tile * 16 + r + half8;
        if (orow < N_NODES) {
            xl[orow * D_HID + col] = accl[r];
            xr[orow * D_HID + col] = accr[r];
        }
    }
}

// ---------------- layer-1 edge scatter: agg1[dst] += xl[src] --------------
// 4 threads per edge, each moves a float4 (16B coalesced) then 4 f32 atomics.
__global__ void sage_scatter1_kernel(const long long* __restrict__ ei,
                                     const float* __restrict__ xl,
                                     float* __restrict__ agg1) {
    int tid = blockIdx.x * blockDim.x + threadIdx.x;
    if (tid >= N_EDGES * 4) return;
    const int e    = tid >> 2;
    const int part = (tid & 3) << 2;                 // 0,4,8,12
    const int s    = (int)ei[e];                     // src row
    const int d    = (int)ei[N_EDGES + e];           // dst row
    const float4 v = *(const float4*)(xl + (size_t)s * D_HID + part);
    float* ap = agg1 + (size_t)d * D_HID + part;
    atomicAdd(ap + 0, v.x);
    atomicAdd(ap + 1, v.y);
    atomicAdd(ap + 2, v.z);
    atomicAdd(ap + 3, v.w);
}

// ---------------- layer-1 epilogue: h = relu(agg1/deg + b1 + xr) ----------
__global__ void sage_finish1_kernel(const float* __restrict__ agg1,
                                    const float* __restrict__ cnt,
                                    const float* __restrict__ b1,
                                    const float* __restrict__ xr,
                                    float* __restrict__ h) {
    int tid = blockIdx.x * blockDim.x + threadIdx.x;
    if (tid >= N_NODES * D_HID) return;
    const int i = tid >> 4;
    const int j = tid & 15;
    const float inv = 1.0f / fmaxf(cnt[i], 1.0f);
    const float v = agg1[tid] * inv + b1[j] + xr[tid];
    h[tid] = fmaxf(v, 0.0f);
}

// ---------------- layer-2 GEMM: hl = h@W2l, hr = h@W2r (WMMA f32) ---------
// K=16 -> 4 WMMA steps; B is 16x8, zero-padded to 16 columns in-register.
__global__ __launch_bounds__(256)
void sage_gemm2_kernel(const float* __restrict__ h,
                       const float* __restrict__ W2l,
                       const float* __restrict__ W2r,
                       float* __restrict__ hl,
                       float* __restrict__ hr) {
    const int lane  = threadIdx.x & 31;
    const int wave  = threadIdx.x >> 5;
    const int tile  = blockIdx.x * 8 + wave;
    if (tile >= NTILES) return;

    const int col   = lane & 15;
    const int khalf = (lane >> 4) << 1;

    int arow = tile * 16 + col;
    if (arow >= N_NODES) arow = N_NODES - 1;
    const float* hrow = h + (size_t)arow * D_HID;

    v8f accl = {0.f,0.f,0.f,0.f,0.f,0.f,0.f,0.f};
    v8f accr = {0.f,0.f,0.f,0.f,0.f,0.f,0.f,0.f};

    for (int k = 0; k < D_HID; k += 4) {
        const int kk = k + khalf;
        v2f a;  a.x = hrow[kk]; a.y = hrow[kk + 1];
        v2f bl = {0.f, 0.f};
        v2f br = {0.f, 0.f};
        if (col < N_CLASSES) {      // zero-pad columns 8..15
            bl.x = W2l[kk * N_CLASSES + col];  bl.y = W2l[(kk + 1) * N_CLASSES + col];
            br.x = W2r[kk * N_CLASSES + col];  br.y = W2r[(kk + 1) * N_CLASSES + col];
        }
        accl = __builtin_amdgcn_wmma_f32_16x16x4_f32(false, a, false, bl,
                                                     (short)0, accl, false, false);
        accr = __builtin_amdgcn_wmma_f32_16x16x4_f32(false, a, false, br,
                                                     (short)0, accr, false, false);
    }

    const int half8 = (lane >> 4) << 3;
#pragma unroll
    for (int r = 0; r < 8; ++r) {
        const int orow = tile * 16 + r + half8;
        if (orow < N_NODES && col < N_CLASSES) {
            hl[orow * N_CLASSES + col] = accl[r];
            hr[orow * N_CLASSES + col] = accr[r];
        }
    }
}

// ---------------- layer-2 edge scatter: agg2[dst] += hl[src] --------------
__global__ void sage_scatter2_kernel(const long long* __restrict__ ei,
                                     const float* __restrict__ hl,
                                     float* __restrict__ agg2) {
    int tid = blockIdx.x * blockDim.x + threadIdx.x;
    if (tid >= N_EDGES * 2) return;
    const int e    = tid >> 1;
    const int part = (tid & 1) << 2;                 // 0 or 4
    const int s    = (int)ei[e];
    const int d    = (int)ei[N_EDGES + e];
    const float4 v = *(const float4*)(hl + (size_t)s * N_CLASSES + part);
    float* ap = agg2 + (size_t)d * N_CLASSES + part;
    atomicAdd(ap + 0, v.x);
    atomicAdd(ap + 1, v.y);
    atomicAdd(ap + 2, v.z);
    atomicAdd(ap + 3, v.w);
}

// ---------------- layer-2 epilogue: out = agg2/deg + b2 + hr --------------
__global__ void sage_finish2_kernel(const float* __restrict__ agg2,
                                    const float* __restrict__ cnt,
                                    const float* __restrict__ b2,
                                    const float* __restrict__ hr,
                                    float* __restrict__ out) {
    int tid = blockIdx.x * blockDim.x + threadIdx.x;
    if (tid >= N_NODES * N_CLASSES) return;
    const int i = tid >> 3;
    const int j = tid & 7;
    const float inv = 1.0f / fmaxf(cnt[i], 1.0f);
    out[tid] = agg2[tid] * inv + b2[j] + hr[tid];
}

// ---------------------------------------------------------------------------
extern "C" void kernel_launch(void* const* d_in, const int* in_sizes, int n_in,
                              void* d_out, int out_size, void* d_ws, size_t ws_size,
                              hipStream_t stream) {
    const float*     x   = (const float*)d_in[0];
    const long long* ei  = (const long long*)d_in[1];   // [2, E] int64
    const float*     W1l = (const float*)d_in[2];
    const float*     b1  = (const float*)d_in[3];
    const float*     W1r = (const float*)d_in[4];
    const float*     W2l = (const float*)d_in[5];
    const float*     b2  = (const float*)d_in[6];
    const float*     W2r = (const float*)d_in[7];
    float* out = (float*)d_out;

    // Workspace carve-up (all 16B-aligned; total ~17.8MB)
    float* ws   = (float*)d_ws;
    float* cnt  = ws;                              // N
    float* xl   = cnt  + N_NODES;                  // N*16
    float* xr   = xl   + N_NODES * D_HID;          // N*16
    float* agg1 = xr   + N_NODES * D_HID;          // N*16
    float* hbuf = agg1 + N_NODES * D_HID;          // N*16
    float* hl   = hbuf + N_NODES * D_HID;          // N*8
    float* hr   = hl   + N_NODES * N_CLASSES;      // N*8
    float* agg2 = hr   + N_NODES * N_CLASSES;      // N*8

    const int B = 256;
    // Re-zero accumulators every call (deterministic; harness may replay).
    sage_zero_kernel<<<(N_NODES + B - 1) / B, B, 0, stream>>>(cnt, N_NODES);
    sage_zero_kernel<<<(N_NODES * D_HID + B - 1) / B, B, 0, stream>>>(agg1, N_NODES * D_HID);
    sage_zero_kernel<<<(N_NODES * N_CLASSES + B - 1) / B, B, 0, stream>>>(agg2, N_NODES * N_CLASSES);

    sage_degree_kernel<<<(N_EDGES + B - 1) / B, B, 0, stream>>>(ei + N_EDGES, cnt);

    sage_gemm1_kernel<<<(NTILES + 7) / 8, B, 0, stream>>>(x, W1l, W1r, xl, xr);
    sage_scatter1_kernel<<<(N_EDGES * 4 + B - 1) / B, B, 0, stream>>>(ei, xl, agg1);
    sage_finish1_kernel<<<(N_NODES * D_HID + B - 1) / B, B, 0, stream>>>(agg1, cnt, b1, xr, hbuf);

    sage_gemm2_kernel<<<(NTILES + 7) / 8, B, 0, stream>>>(hbuf, W2l, W2r, hl, hr);
    sage_scatter2_kernel<<<(N_EDGES * 2 + B - 1) / B, B, 0, stream>>>(ei, hl, agg2);
    sage_finish2_kernel<<<(N_NODES * N_CLASSES + B - 1) / B, B, 0, stream>>>(agg2, cnt, b2, hr, out);
}